// RNNLM_40630390620357
// MI455X (gfx1250) — compile-verified
//
#include <hip/hip_runtime.h>
#include <hip/hip_bf16.h>
#include <math.h>

// Sizes from the reference
#define VOCAB  32000
#define EMBED  32
#define HIDDEN 16
#define SEQ    64
#define BATCH  64

// WMMA fragment types for V_WMMA_F32_16X16X4_F32 (wave32):
//   A: 16x4 f32  -> 2 VGPRs  (v2f)
//   B: 4x16 f32  -> 2 VGPRs  (v2f)
//   C/D: 16x16   -> 8 VGPRs  (v8f)
typedef float v2f __attribute__((ext_vector_type(2)));
typedef float v8f __attribute__((ext_vector_type(8)));

// -------------------------------------------------------------------------
// Phase 1: serial RNN recurrence. One workgroup; h state in LDS.
// Writes h_all[t*B*H + b*H + j] and zeroes ws_sum[t].
// -------------------------------------------------------------------------
__global__ __launch_bounds__(256)
void rnnlm_hidden_kernel(const int* __restrict__ inp,
                         const float* __restrict__ hidden0,
                         const float* __restrict__ we,
                         const float* __restrict__ i2h,
                         float* __restrict__ h_all,
                         double* __restrict__ ws_sum) {
    __shared__ float s_i2h[(EMBED + HIDDEN) * HIDDEN];   // 48x16
    __shared__ float s_h[2][BATCH * HIDDEN];             // double-buffered state
    const int tid = threadIdx.x;                         // 256 threads

    for (int i = tid; i < (EMBED + HIDDEN) * HIDDEN; i += 256) s_i2h[i] = i2h[i];
    for (int i = tid; i < BATCH * HIDDEN; i += 256)            s_h[0][i] = hidden0[i];
    if (tid < SEQ) ws_sum[tid] = 0.0;                    // fresh every launch
    __syncthreads();

    const int b  = tid >> 2;          // batch row this thread owns
    const int j0 = (tid & 3) << 2;    // 4 hidden outputs per thread
    int cur = 0;

    for (int t = 0; t < SEQ; ++t) {
        const int   idx = inp[t * BATCH + b];
        const float* emb = we + (size_t)idx * EMBED;

        float acc[4] = {0.f, 0.f, 0.f, 0.f};
        #pragma unroll 4
        for (int e = 0; e < EMBED; ++e) {
            const float ev = emb[e];
            #pragma unroll
            for (int jj = 0; jj < 4; ++jj) acc[jj] += ev * s_i2h[e * HIDDEN + j0 + jj];
        }
        #pragma unroll
        for (int k = 0; k < HIDDEN; ++k) {
            const float hv = s_h[cur][b * HIDDEN + k];
            #pragma unroll
            for (int jj = 0; jj < 4; ++jj) acc[jj] += hv * s_i2h[(EMBED + k) * HIDDEN + j0 + jj];
        }
        #pragma unroll
        for (int jj = 0; jj < 4; ++jj) {
            const float hn = tanhf(acc[jj]);
            s_h[cur ^ 1][b * HIDDEN + j0 + jj] = hn;
            h_all[(size_t)t * BATCH * HIDDEN + b * HIDDEN + j0 + jj] = hn;
        }
        cur ^= 1;
        __syncthreads();
    }
}

// -------------------------------------------------------------------------
// Shared WMMA tile helper: computes one 16x16 f32 tile of H_t @ i2o
// using 4 chained v_wmma_f32_16x16x4_f32 (K = 16).
//   afrag[kk] holds A(16x4) for K-slice kk (hoisted by caller, reused).
//   Wp = &i2o[ncol + (lane&15)]; K rows stride VOCAB.
// -------------------------------------------------------------------------
__device__ __forceinline__
v8f wmma_tile_f32(const v2f afrag[4], const float* __restrict__ Wp, int khalf) {
    v8f acc = {};
    #pragma unroll
    for (int kk = 0; kk < 4; ++kk) {
        v2f bfrag;
        bfrag.x = Wp[(size_t)(4 * kk + khalf)     * VOCAB];
        bfrag.y = Wp[(size_t)(4 * kk + khalf + 1) * VOCAB];
        acc = __builtin_amdgcn_wmma_f32_16x16x4_f32(
                  /*neg_a=*/false, afrag[kk],
                  /*neg_b=*/false, bfrag,
                  /*c_mod=*/(short)0, acc,
                  /*reuse_a=*/false, /*reuse_b=*/false);
    }
    return acc;
}

// Grid decomposition for the big GEMM passes:
//   block = 256 threads = 8 waves; each wave does 5 consecutive 16-col tiles.
//   8 waves * 5 tiles * 16 cols = 640 cols/block; 50 vchunks cover VOCAB=32000.
//   blockIdx.x encodes (t, btile, vchunk): 64 * 4 * 50 = 12800 blocks.
#define VCHUNKS 50
#define TILES_PER_WAVE 5

// -------------------------------------------------------------------------
// Pass B: recompute logits with WMMA, accumulate sum(exp(logit)) per t in f64.
// -------------------------------------------------------------------------
__global__ __launch_bounds__(256)
void rnnlm_sumexp_kernel(const float* __restrict__ h_all,
                         const float* __restrict__ i2o,
                         double* __restrict__ ws_sum) {
    const int t      = blockIdx.x / (4 * VCHUNKS);
    const int rem    = blockIdx.x % (4 * VCHUNKS);
    const int btile  = rem / VCHUNKS;
    const int vchunk = rem % VCHUNKS;

    const int lane  = threadIdx.x & 31;
    const int wave  = threadIdx.x >> 5;
    const int row   = lane & 15;            // M for A, N for B
    const int khalf = (lane >> 4) << 1;     // K sub-slice: {0,1} or {2,3}

    // Hoisted A fragments (16x4 per K-slice), reused across all N tiles.
    const float* Hp = h_all + (size_t)t * BATCH * HIDDEN + (btile * 16 + row) * HIDDEN;
    v2f afrag[4];
    #pragma unroll
    for (int kk = 0; kk < 4; ++kk) {
        afrag[kk].x = Hp[4 * kk + khalf];
        afrag[kk].y = Hp[4 * kk + khalf + 1];
    }

    double lsum = 0.0;
    #pragma unroll
    for (int it = 0; it < TILES_PER_WAVE; ++it) {
        const int ncol = (vchunk * (8 * TILES_PER_WAVE) + wave * TILES_PER_WAVE + it) * 16;
        const float* Wp = i2o + (size_t)ncol + row;
        v8f acc = wmma_tile_f32(afrag, Wp, khalf);
        #pragma unroll
        for (int g = 0; g < 8; ++g) lsum += (double)__expf(acc[g]);
    }

    // Block-level f64 reduction, then one f64 atomic per block into ws_sum[t].
    __shared__ double sred[256];
    sred[threadIdx.x] = lsum;
    __syncthreads();
    #pragma unroll
    for (int s = 128; s > 0; s >>= 1) {
        if ((int)threadIdx.x < s) sred[threadIdx.x] += sred[threadIdx.x + s];
        __syncthreads();
    }
    if (threadIdx.x == 0) atomicAdd(&ws_sum[t], sred[0]);   // global_atomic_add_f64
}

// -------------------------------------------------------------------------
// Pass C: recompute logits with WMMA, subtract log(sum), stream 512 MB out
// with non-temporal stores (pure streaming write; the HBM floor ~22 us).
// -------------------------------------------------------------------------
__global__ __launch_bounds__(256)
void rnnlm_output_kernel(const float* __restrict__ h_all,
                         const float* __restrict__ i2o,
                         const double* __restrict__ ws_sum,
                         float* __restrict__ out) {
    const int t      = blockIdx.x / (4 * VCHUNKS);
    const int rem    = blockIdx.x % (4 * VCHUNKS);
    const int btile  = rem / VCHUNKS;
    const int vchunk = rem % VCHUNKS;

    const int lane  = threadIdx.x & 31;
    const int wave  = threadIdx.x >> 5;
    const int row   = lane & 15;
    const int khalf = (lane >> 4) << 1;

    const float lse = (float)log(ws_sum[t]);   // per-t global logsumexp (no-shift, f64 sum)

    const float* Hp = h_all + (size_t)t * BATCH * HIDDEN + (btile * 16 + row) * HIDDEN;
    v2f afrag[4];
    #pragma unroll
    for (int kk = 0; kk < 4; ++kk) {
        afrag[kk].x = Hp[4 * kk + khalf];
        afrag[kk].y = Hp[4 * kk + khalf + 1];
    }

    const int mhi = (lane >> 4) << 3;          // lanes 16-31 hold rows M = g + 8
    const int ncl = lane & 15;                 // N within tile

    #pragma unroll
    for (int it = 0; it < TILES_PER_WAVE; ++it) {
        const int ncol = (vchunk * (8 * TILES_PER_WAVE) + wave * TILES_PER_WAVE + it) * 16;
        const float* Wp = i2o + (size_t)ncol + row;
        v8f acc = wmma_tile_f32(afrag, Wp, khalf);
        #pragma unroll
        for (int g = 0; g < 8; ++g) {
            const int M = g + mhi;
            const size_t o = ((size_t)t * BATCH + (btile * 16 + M)) * VOCAB + ncol + ncl;
            __builtin_nontemporal_store(acc[g] - lse, &out[o]);
        }
    }
}

// -------------------------------------------------------------------------
// Launch: inputs in setup_inputs() order: inp(i32), hidden0, we, i2h, i2o.
// Workspace: h_all (256 KB f32) + ws_sum (64 f64).
// -------------------------------------------------------------------------
extern "C" void kernel_launch(void* const* d_in, const int* in_sizes, int n_in,
                              void* d_out, int out_size, void* d_ws, size_t ws_size,
                              hipStream_t stream) {
    const int*   inp     = (const int*)  d_in[0];
    const float* hidden0 = (const float*)d_in[1];
    const float* we      = (const float*)d_in[2];
    const float* i2h     = (const float*)d_in[3];
    const float* i2o     = (const float*)d_in[4];
    float*       out     = (float*)d_out;

    float*  h_all  = (float*)d_ws;                                        // 65536 f32
    double* ws_sum = (double*)((char*)d_ws + (size_t)SEQ * BATCH * HIDDEN * sizeof(float));

    rnnlm_hidden_kernel<<<1, 256, 0, stream>>>(inp, hidden0, we, i2h, h_all, ws_sum);

    const int nblocks = SEQ * 4 * VCHUNKS;   // 12800
    rnnlm_sumexp_kernel<<<nblocks, 256, 0, stream>>>(h_all, i2o, ws_sum);
    rnnlm_output_kernel<<<nblocks, 256, 0, stream>>>(h_all, i2o, ws_sum, out);
}